// TILDEQ_loss_56298431316512
// MI455X (gfx1250) — compile-verified
//
#include <hip/hip_runtime.h>
#include <stdint.h>

// Problem constants (from the reference)
#define B_ROWS 8192
#define TI     1440
#define TO     720
#define BLK    256   // 8 waves of 32

typedef float v2f __attribute__((ext_vector_type(2)));
typedef float v8f __attribute__((ext_vector_type(8)));

// ---- CDNA5 async global->LDS helpers -------------------------------------
// the LDS byte address is taken from the vdst VGPR plus the offset; generic shared pointers carry the
// workgroup-relative LDS byte offset in their low 32 bits.
__device__ __forceinline__ uint32_t lds_off(const void* p) {
  return (uint32_t)(uintptr_t)p;
}

__device__ __forceinline__ void async_ld16(uint32_t lds_byte, const float* g) {
  uint64_t ga = (uint64_t)(uintptr_t)g;
  asm volatile("global_load_async_to_lds_b128 %0, %1, off"
               :: "v"(lds_byte), "v"(ga) : "memory");
}

__device__ __forceinline__ void wait_async0() {
  asm volatile("s_wait_asynccnt 0" ::: "memory");
}

// ---- block reductions (wave32 shfl + LDS across 8 waves) -----------------
__device__ __forceinline__ float block_sum(float v, float* red) {
  __syncthreads();                       // red reuse safety
  int lane = threadIdx.x & 31, w = threadIdx.x >> 5;
  #pragma unroll
  for (int o = 16; o > 0; o >>= 1) v += __shfl_down(v, o, 32);
  if (lane == 0) red[w] = v;
  __syncthreads();
  if (threadIdx.x == 0) {
    float r = 0.f;
    #pragma unroll
    for (int i = 0; i < BLK / 32; ++i) r += red[i];
    red[8] = r;
  }
  __syncthreads();
  return red[8];
}

__device__ __forceinline__ float block_max(float v, float* red) {
  __syncthreads();
  int lane = threadIdx.x & 31, w = threadIdx.x >> 5;
  #pragma unroll
  for (int o = 16; o > 0; o >>= 1) v = fmaxf(v, __shfl_down(v, o, 32));
  if (lane == 0) red[w] = v;
  __syncthreads();
  if (threadIdx.x == 0) {
    float r = red[0];
    #pragma unroll
    for (int i = 1; i < BLK / 32; ++i) r = fmaxf(r, red[i]);
    red[8] = r;
  }
  __syncthreads();
  return red[8];
}

// ---- Kernel A: one workgroup per row; single HBM pass via async LDS ------
__global__ __launch_bounds__(BLK)
void tildeq_row_kernel(const float* __restrict__ ins,
                       const int*   __restrict__ freqp,
                       const float* __restrict__ fc,
                       const float* __restrict__ tg,
                       const float* __restrict__ mk,
                       float* __restrict__ part) {
  __shared__ __align__(16) float sF[TO];
  __shared__ __align__(16) float sT[TO];
  __shared__ __align__(16) float sM[TO];
  __shared__ __align__(16) float sI[TI];
  __shared__ float red[9];

  const int row  = blockIdx.x;
  const int tid  = threadIdx.x;
  const int freq = *freqp;

  const float* fRow = fc  + (size_t)row * TO;
  const float* tRow = tg  + (size_t)row * TO;
  const float* mRow = mk  + (size_t)row * TO;
  const float* iRow = ins + (size_t)row * TI;

  // Stage rows into LDS with 128-bit async copies (read HBM exactly once).
  if (tid < TO / 4) {                    // 180 x 16B per 720-float row
    uint32_t o = (uint32_t)tid * 16u;
    async_ld16(lds_off(sF) + o, fRow + tid * 4);
    async_ld16(lds_off(sT) + o, tRow + tid * 4);
    async_ld16(lds_off(sM) + o, mRow + tid * 4);
  }
  for (int c = tid; c < TI / 4; c += BLK)          // 360 x 16B insample row
    async_ld16(lds_off(sI) + (uint32_t)c * 16u, iRow + c * 4);
  wait_async0();
  __syncthreads();

  // --- softmax term: dyy = (t - f) * m ; ashift = T * sum|1/T - softmax| ---
  float d0[3];
  float lmax = -INFINITY;
  #pragma unroll
  for (int k = 0; k < 3; ++k) {
    int i = tid + k * BLK;
    if (i < TO) {
      float d = (sT[i] - sF[i]) * sM[i];
      d0[k] = d;
      lmax = fmaxf(lmax, d);
    } else d0[k] = -INFINITY;
  }
  float rmax = block_max(lmax, red);

  float lsum = 0.f;
  #pragma unroll
  for (int k = 0; k < 3; ++k) {
    int i = tid + k * BLK;
    if (i < TO) {
      float e = expf(d0[k] - rmax);
      d0[k] = e;
      lsum += e;
    }
  }
  float Z    = block_sum(lsum, red);
  float invZ = 1.f / Z;
  const float invT = 1.f / (float)TO;

  float leq = 0.f, lsm = 0.f, lmase = 0.f;
  #pragma unroll
  for (int k = 0; k < 3; ++k) {
    int i = tid + k * BLK;
    if (i < TO) {
      leq += fabsf(invT - d0[k] * invZ);
      float fv = sF[i], tv = sT[i], mv = sM[i];
      float ad  = fabsf(fv - tv);
      float den = fabsf(fv) + fabsf(tv);
      lsm   += (den > 0.f ? ad / den : 0.f) * mv;   // divide_no_nan
      lmase += ad * mv;
    }
  }
  float eqsum   = block_sum(leq,   red);
  float smsum   = block_sum(lsm,   red);
  float masesum = block_sum(lmase, red);

  // --- MASE denominator: mean |ins[i+freq] - ins[i]| over T_I - freq ------
  float lmp = 0.f;
  const int L = TI - freq;
  for (int i = tid; i < L; i += BLK) lmp += fabsf(sI[i + freq] - sI[i]);
  float mpsum = block_sum(lmp, red);

  if (tid == 0) {
    float masep = mpsum / (float)L;
    part[row]              = 0.99f * (float)TO * eqsum;            // TILDEQ
    part[B_ROWS + row]     = smsum;                                // sMAPE num
    part[2 * B_ROWS + row] = (masep > 0.f) ? masesum / masep : 0.f; // divide_no_nan
  }
}

// ---- Kernel B: one wave; WMMA f32 16x16x4 against ones reduces 8192 ------
__global__ __launch_bounds__(32)
void tildeq_final_kernel(const float* __restrict__ part, float* __restrict__ out) {
  const int lane = threadIdx.x;
  float tot[3];
  #pragma unroll
  for (int s = 0; s < 3; ++s) {
    const float* p = part + (size_t)s * B_ROWS;
    v8f c = {};
    v2f onesv; onesv[0] = 1.f; onesv[1] = 1.f;
    // D[m,n] = sum_k A[m,k] * 1 + C[m,n]; 64 values folded per wmma,
    // deterministic fixed-order accumulation in C.
    for (int it = 0; it < B_ROWS / 64; ++it) {
      int base = it * 64 + lane * 2;
      v2f a; a[0] = p[base]; a[1] = p[base + 1];
      c = __builtin_amdgcn_wmma_f32_16x16x4_f32(
          /*neg_a=*/false, a, /*neg_b=*/false, onesv,
          /*c_mod=*/(short)0, c, /*reuse_a=*/false, /*reuse_b=*/false);
    }
    // lane n holds sum over M=0..7 of column n; lane n+16 holds M=8..15.
    float s8 = c[0] + c[1] + c[2] + c[3] + c[4] + c[5] + c[6] + c[7];
    tot[s] = __shfl(s8, 0, 32) + __shfl(s8, 16, 32);
  }
  if (lane == 0) {
    const float N = (float)B_ROWS * (float)TO;
    float termA = tot[0] / (float)B_ROWS / 4.f;   // mean(0.99*ashift)/4
    float termS = 200.f * tot[1] / N;             // 200 * smape
    float termM = tot[2] / N;                     // MASE term
    out[0] = termA + termS + termM;
  }
}

extern "C" void kernel_launch(void* const* d_in, const int* in_sizes, int n_in,
                              void* d_out, int out_size, void* d_ws, size_t ws_size,
                              hipStream_t stream) {
  const float* ins  = (const float*)d_in[0];
  const int*   freq = (const int*)  d_in[1];
  const float* fc   = (const float*)d_in[2];
  const float* tg   = (const float*)d_in[3];
  const float* mk   = (const float*)d_in[4];
  float* part = (float*)d_ws;   // 3 * 8192 floats = 96 KB of scratch

  tildeq_row_kernel<<<B_ROWS, BLK, 0, stream>>>(ins, freq, fc, tg, mk, part);
  tildeq_final_kernel<<<1, 32, 0, stream>>>(part, (float*)d_out);
}